// AbAgNet_78993038508487
// MI455X (gfx1250) — compile-verified
//
#include <hip/hip_runtime.h>
#include <hip/hip_bf16.h>
#include <stdint.h>

// Problem constants (from reference)
#define BB   8
#define CC   32
#define GG   2048
#define DD   128
#define DDP  (DD + 4)        // padded LDS row stride (528B rows: 16B aligned, conflict-free)
#define N_AB (BB*CC)         // 256
#define N_AG (BB*GG)         // 16384
#define NT   (N_AB+N_AG)     // 16640
#define SLOPE 0.2f

typedef float v2f __attribute__((ext_vector_type(2)));
typedef float v8f __attribute__((ext_vector_type(8)));
typedef unsigned int u32x4 __attribute__((ext_vector_type(4)));
typedef int i32x4 __attribute__((ext_vector_type(4)));
typedef int i32x8 __attribute__((ext_vector_type(8)));

#if defined(__has_builtin)
# if __has_builtin(__builtin_amdgcn_tensor_load_to_lds)
#  define HAVE_TDM 1
# endif
# if __has_builtin(__builtin_amdgcn_s_wait_tensorcnt)
#  define HAVE_WTC 1
# endif
# if __has_builtin(__builtin_amdgcn_s_wait_asynccnt)
#  define HAVE_WAC 1
# endif
#endif

static __device__ __forceinline__ float lrelu(float x) {
    return x > 0.f ? x : SLOPE * x;
}

static __device__ __forceinline__ v8f wmma_f32(v2f a, v2f b, v8f c) {
    // V_WMMA_F32_16X16X4_F32 : D[16x16] = A[16x4] * B[4x16] + C
    return __builtin_amdgcn_wmma_f32_16x16x4_f32(
        false, a, false, b, (short)0, c, false, false);
}

// --- async global->LDS 16B copy (ASYNCcnt tracked, no VGPR round trip) -----
static __device__ __forceinline__ void async_ld16(float* lds_dst, const float* gsrc) {
    unsigned l = (unsigned)(uintptr_t)lds_dst;                 // LDS byte address
    unsigned long long g = (unsigned long long)(uintptr_t)gsrc;
    asm volatile("global_load_async_to_lds_b128 %0, %1, off"
                 :: "v"(l), "v"(g) : "memory");
}

static __device__ __forceinline__ void wait_async0() {
#if defined(HAVE_WAC)
    __builtin_amdgcn_s_wait_asynccnt(0);
#else
    asm volatile("s_wait_asynccnt 0x0" ::: "memory");
#endif
}

static __device__ __forceinline__ void wait_tensor0() {
#if defined(HAVE_WTC)
    __builtin_amdgcn_s_wait_tensorcnt(0);
#else
    asm volatile("s_wait_tensorcnt 0x0" ::: "memory");
#endif
}

// --- TDM 2D tile load: rows x cols fp32, row stride `stride` elements ------
static __device__ __forceinline__ void tdm_load_2d(float* lds_dst, const float* gptr,
                                                   unsigned cols, unsigned rows,
                                                   unsigned stride) {
#if defined(HAVE_TDM)
    unsigned long long ga = (unsigned long long)(uintptr_t)gptr;
    unsigned lds_off = (unsigned)(uintptr_t)lds_dst;
    u32x4 g0;
    g0[0] = 1u;                                            // count=1, user mode
    g0[1] = lds_off;                                       // lds_addr
    g0[2] = (unsigned)ga;                                  // global_addr[31:0]
    g0[3] = (unsigned)((ga >> 32) & 0x01ffffffu) | 0x80000000u; // ga[56:32] | type=2
    i32x8 g1;
    g1[0] = 0x00020000;                                    // data_size=4B, no mask/pad
    g1[1] = (int)((cols & 0xffffu) << 16);                 // tensor_dim0[15:0] @ [63:48]
    g1[2] = (int)(((cols >> 16) & 0xffffu) | ((rows & 0xffffu) << 16)); // dim0 hi | dim1 lo
    g1[3] = (int)(((rows >> 16) & 0xffffu) | ((cols & 0xffffu) << 16)); // dim1 hi | tile_dim0
    g1[4] = (int)(rows & 0xffffu);                         // tile_dim1 = rows, tile_dim2 = 0
    g1[5] = (int)stride;                                   // tensor_dim0_stride[31:0]
    g1[6] = 0;
    g1[7] = 0;
    i32x4 z4 = {0, 0, 0, 0};
    i32x8 z8 = {0, 0, 0, 0, 0, 0, 0, 0};
    // 6-arg form (clang-23 / therock-10.0 headers): groups 0..3 + extra group + cpol
    __builtin_amdgcn_tensor_load_to_lds(g0, g1, z4, z4, z8, 0);
#else
    // fallback: cooperative async copies (rows*cols must be multiple of 4*256)
    for (int t = threadIdx.x; t < (int)(rows * cols) / 4; t += 256)
        async_ld16(lds_dst + t * 4, gptr + t * 4);
#endif
}

// ---------------------------------------------------------------------------
// h = X @ W   (X virtually concatenated: xa = rows [0,256), xb = rows [256,NT))
// grid = NT/16 blocks x 256 threads (8 waves); wave w -> column tile w (16 cols)
// W staged in LDS via TDM; X stripe staged via async b128 into padded tile.
// ---------------------------------------------------------------------------
__global__ void k_gemm_xw(const float* __restrict__ xa, const float* __restrict__ xb,
                          const float* __restrict__ W, float* __restrict__ h) {
    __shared__ float sW[DD * DD];      // 64 KB
    __shared__ float sX[16 * DDP];     // 8.25 KB, padded rows
    const int wave = threadIdx.x >> 5;
    const int lane = threadIdx.x & 31;
    const int row0 = blockIdx.x * 16;
    const int n0 = wave * 16;
    const int m  = lane & 15;
    const int kh = lane >> 4;          // 0: K=k0,k0+1   1: K=k0+2,k0+3

    // stage X stripe (16 rows x 128): 512 16B-transfers, 2 per thread
    const float* xsrc = (row0 < N_AB) ? (xa + (size_t)row0 * DD)
                                      : (xb + (size_t)(row0 - N_AB) * DD);
#pragma unroll
    for (int u = 0; u < 2; ++u) {
        const int idx = threadIdx.x + u * 256;   // float4 index into [16][128]
        const int rr = idx >> 5;                 // row (32 float4 per row)
        const int c4 = idx & 31;
        async_ld16(&sX[rr * DDP + c4 * 4], xsrc + idx * 4);
    }
    // stage W (128x128) via Tensor Data Mover from wave 0
#if defined(HAVE_TDM)
    if (wave == 0) {
        tdm_load_2d(sW, W, DD, DD, DD);
        wait_tensor0();
    }
#else
    tdm_load_2d(sW, W, DD, DD, DD);    // async-copy fallback (all threads)
#endif
    wait_async0();
    __syncthreads();

    v8f acc = {};
#pragma unroll 4
    for (int k0 = 0; k0 < DD; k0 += 4) {
        const int ka = k0 + 2 * kh;
        v2f a, b;
        a[0] = sX[m * DDP + ka];
        a[1] = sX[m * DDP + ka + 1];
        b[0] = sW[ka * DD + n0 + m];
        b[1] = sW[(ka + 1) * DD + n0 + m];
        acc = wmma_f32(a, b, acc);
    }
#pragma unroll
    for (int rr = 0; rr < 8; ++rr) {
        const int ro = row0 + rr + 8 * kh;
        h[(size_t)ro * DD + n0 + m] = acc[rr];
    }
}

// ---------------------------------------------------------------------------
// al_s = h @ a_s ; al_d = h @ a_d   (one wave per row)
// ---------------------------------------------------------------------------
__global__ void k_dots(const float* __restrict__ h, const float* __restrict__ as,
                       const float* __restrict__ ad, float* __restrict__ als,
                       float* __restrict__ ald) {
    const int wave = threadIdx.x >> 5;
    const int lane = threadIdx.x & 31;
    const int row = blockIdx.x * 8 + wave;
    if (row >= NT) return;
    float s = 0.f, d = 0.f;
    for (int k = lane; k < DD; k += 32) {
        const float hv = h[(size_t)row * DD + k];
        s += hv * as[k];
        d += hv * ad[k];
    }
#pragma unroll
    for (int off = 16; off; off >>= 1) {
        s += __shfl_xor(s, off);
        d += __shfl_xor(d, off);
    }
    if (lane == 0) { als[row] = s; ald[row] = d; }
}

// ---------------------------------------------------------------------------
// Softmax stats (max, sum-exp) for AB destination nodes: 2048 AG edges + self
// ---------------------------------------------------------------------------
__global__ void k_ab_stats(const float* __restrict__ als, const float* __restrict__ ald,
                           float* __restrict__ mmax, float* __restrict__ zsum) {
    const int wave = threadIdx.x >> 5;
    const int lane = threadIdx.x & 31;
    const int i = blockIdx.x * 8 + wave;        // AB node id, 0..255
    const int b = i >> 5;
    const int agbase = N_AB + b * GG;
    const float ad = ald[i];
    const float eself = lrelu(als[i] + ad);
    float mx = eself;
    for (int j = lane; j < GG; j += 32)
        mx = fmaxf(mx, lrelu(als[agbase + j] + ad));
#pragma unroll
    for (int off = 16; off; off >>= 1) mx = fmaxf(mx, __shfl_xor(mx, off));
    float z = 0.f;
    for (int j = lane; j < GG; j += 32)
        z += __expf(lrelu(als[agbase + j] + ad) - mx);
#pragma unroll
    for (int off = 16; off; off >>= 1) z += __shfl_xor(z, off);
    z += __expf(eself - mx);
    if (lane == 0) { mmax[i] = mx; zsum[i] = z; }
}

// ---------------------------------------------------------------------------
// AB aggregation: out_ab[32,128] = alpha[32,2048] @ h_ag[2048,128] (per batch)
//                 + alpha_self * h_ab + bias  (+ relu for layer 1)
// h_ag streamed through LDS in 64-row chunks via async b128 copies.
// grid = 8 batches * 2 row-tiles; wave = column tile
// ---------------------------------------------------------------------------
#define KC 64
__global__ void k_ab_agg(const float* __restrict__ h, const float* __restrict__ als,
                         const float* __restrict__ ald, const float* __restrict__ mmax,
                         const float* __restrict__ zsum, const float* __restrict__ bias,
                         float* __restrict__ out, int do_relu) {
    __shared__ float sh[KC * DD];      // 32 KB chunk of h_ag
    const int wave = threadIdx.x >> 5;
    const int lane = threadIdx.x & 31;
    const int bb = blockIdx.x >> 1;
    const int mt = blockIdx.x & 1;
    const int n0 = wave * 16;
    const int m  = lane & 15;
    const int kh = lane >> 4;
    const int i = bb * 32 + mt * 16 + m;        // AB node for this A-row
    const int agbase = N_AB + bb * GG;
    const float* hag = h + (size_t)agbase * DD;
    const float ad = ald[i];
    const float mx = mmax[i];
    const float iz = 1.f / (zsum[i] + 1e-16f);
    v8f acc = {};
    for (int kc = 0; kc < GG; kc += KC) {
        // cooperative stage of rows [kc, kc+KC): 2048 float4s, 8 per thread
#pragma unroll
        for (int u = 0; u < (KC * DD / 4) / 256; ++u) {
            const int idx = threadIdx.x + u * 256;
            async_ld16(&sh[idx * 4], hag + (size_t)kc * DD + idx * 4);
        }
        wait_async0();
        __syncthreads();
#pragma unroll 4
        for (int k0 = 0; k0 < KC; k0 += 4) {
            const int k = k0 + 2 * kh;
            const int kg = kc + k;
            v2f a, bf;
            a[0] = __expf(lrelu(als[agbase + kg] + ad) - mx) * iz;
            a[1] = __expf(lrelu(als[agbase + kg + 1] + ad) - mx) * iz;
            bf[0] = sh[k * DD + n0 + m];
            bf[1] = sh[(k + 1) * DD + n0 + m];
            acc = wmma_f32(a, bf, acc);
        }
        __syncthreads();
    }
    const int col = n0 + m;
#pragma unroll
    for (int rr = 0; rr < 8; ++rr) {
        const int ro = bb * 32 + mt * 16 + rr + 8 * kh;   // AB node of this C row
        const float mxo = mmax[ro];
        const float izo = 1.f / (zsum[ro] + 1e-16f);
        const float aself = __expf(lrelu(als[ro] + ald[ro]) - mxo) * izo;
        float val = acc[rr] + aself * h[(size_t)ro * DD + col] + bias[col];
        if (do_relu) val = fmaxf(val, 0.f);
        out[(size_t)ro * DD + col] = val;
    }
}

// ---------------------------------------------------------------------------
// AG aggregation: out_ag[2048,128] = alpha[2048,32] @ h_ab[32,128] (per batch)
//                 + alpha_self * h_ag + bias  (+ relu for layer 1)
// h_ab block (16 KB) staged in LDS via async copies.
// ---------------------------------------------------------------------------
__global__ void k_ag_agg(const float* __restrict__ h, const float* __restrict__ als,
                         const float* __restrict__ ald, const float* __restrict__ bias,
                         float* __restrict__ out, int do_relu) {
    __shared__ float shb[32 * DD];     // 16 KB: this batch's h_ab
    __shared__ float smx[16], siz[16];
    const int wave = threadIdx.x >> 5;
    const int lane = threadIdx.x & 31;
    const int bb = blockIdx.x >> 7;             // 128 stripes per batch
    const int mt = blockIdx.x & 127;
    const int n0 = wave * 16;
    const int m  = lane & 15;
    const int kh = lane >> 4;
    const int node = N_AB + bb * GG + mt * 16 + m;  // AG node for this A-row
    const int abbase = bb * 32;

    // stage h_ab[32][128]: 1024 float4s, 4 per thread
#pragma unroll
    for (int u = 0; u < 4; ++u) {
        const int idx = threadIdx.x + u * 256;
        async_ld16(&shb[idx * 4], h + (size_t)abbase * DD + idx * 4);
    }

    const float ad = ald[node];
    const float eself = lrelu(als[node] + ad);
    float mx = eself;
#pragma unroll 8
    for (int i2 = 0; i2 < 32; ++i2)
        mx = fmaxf(mx, lrelu(als[abbase + i2] + ad));
    float z = __expf(eself - mx);
#pragma unroll 8
    for (int i2 = 0; i2 < 32; ++i2)
        z += __expf(lrelu(als[abbase + i2] + ad) - mx);
    const float iz = 1.f / (z + 1e-16f);
    if (wave == 0 && kh == 0) { smx[m] = mx; siz[m] = iz; }
    wait_async0();
    __syncthreads();

    v8f acc = {};
#pragma unroll
    for (int k0 = 0; k0 < 32; k0 += 4) {
        const int k = k0 + 2 * kh;
        v2f a, bf;
        a[0] = __expf(lrelu(als[abbase + k] + ad) - mx) * iz;
        a[1] = __expf(lrelu(als[abbase + k + 1] + ad) - mx) * iz;
        bf[0] = shb[k * DD + n0 + m];
        bf[1] = shb[(k + 1) * DD + n0 + m];
        acc = wmma_f32(a, bf, acc);
    }
    const int col = n0 + m;
#pragma unroll
    for (int rr = 0; rr < 8; ++rr) {
        const int rloc = rr + 8 * kh;
        const int ro = N_AB + bb * GG + mt * 16 + rloc;
        const float aself = __expf(lrelu(als[ro] + ald[ro]) - smx[rloc]) * siz[rloc];
        float val = acc[rr] + aself * h[(size_t)ro * DD + col] + bias[col];
        if (do_relu) val = fmaxf(val, 0.f);
        out[(size_t)ro * DD + col] = val;
    }
}

// ---------------------------------------------------------------------------
// BatchNorm stats, AB branch: concat(x2_ab, selected_ab) -> [256, 256]
// ---------------------------------------------------------------------------
__global__ void k_bn_ab(const float* __restrict__ x2, const float* __restrict__ sab,
                        float* __restrict__ mean, float* __restrict__ var) {
    const int c = threadIdx.x;    // 0..255
    float s = 0.f, ss = 0.f;
    for (int r = 0; r < N_AB; ++r) {
        const float v = (c < DD) ? x2[(size_t)r * DD + c]
                                 : sab[(size_t)r * DD + c - DD];
        s += v; ss += v * v;
    }
    const float mn = s * (1.f / N_AB);
    mean[c] = mn;
    var[c]  = ss * (1.f / N_AB) - mn * mn;
}

// ---------------------------------------------------------------------------
// BatchNorm stats, AG branch: concat(x2_ag, x_ag) -> [16384, 256]
// ---------------------------------------------------------------------------
__global__ void k_bn_ag(const float* __restrict__ x2, const float* __restrict__ xag,
                        float* __restrict__ mean, float* __restrict__ var) {
    __shared__ float ssum[256], ssq[256];
    const int c = blockIdx.x;     // 0..255
    float s = 0.f, ss = 0.f;
    for (int r = threadIdx.x; r < N_AG; r += 256) {
        const float v = (c < DD) ? x2[(size_t)(N_AB + r) * DD + c]
                                 : xag[(size_t)r * DD + c - DD];
        s += v; ss += v * v;
    }
    ssum[threadIdx.x] = s; ssq[threadIdx.x] = ss;
    __syncthreads();
    for (int off = 128; off; off >>= 1) {
        if (threadIdx.x < off) {
            ssum[threadIdx.x] += ssum[threadIdx.x + off];
            ssq[threadIdx.x]  += ssq[threadIdx.x + off];
        }
        __syncthreads();
    }
    if (threadIdx.x == 0) {
        const float mn = ssum[0] * (1.f / N_AG);
        mean[c] = mn;
        var[c]  = ssq[0] * (1.f / N_AG) - mn * mn;
    }
}

// ---------------------------------------------------------------------------
// Heads: per row, BN-normalize, relu, dot with fc weights. one wave per row.
// ---------------------------------------------------------------------------
__global__ void k_head(const float* __restrict__ x2, const float* __restrict__ sab,
                       const float* __restrict__ xag,
                       const float* __restrict__ mab, const float* __restrict__ vab,
                       const float* __restrict__ gab, const float* __restrict__ bab,
                       const float* __restrict__ mag, const float* __restrict__ vag,
                       const float* __restrict__ gag, const float* __restrict__ bag,
                       const float* __restrict__ fcw, const float* __restrict__ fcb,
                       const float* __restrict__ agw, const float* __restrict__ agb,
                       float* __restrict__ out) {
    const int wave = threadIdx.x >> 5;
    const int lane = threadIdx.x & 31;
    const int row = blockIdx.x * 8 + wave;
    if (row >= NT) return;
    const bool isab = row < N_AB;
    const float* mean = isab ? mab : mag;
    const float* var  = isab ? vab : vag;
    const float* gam  = isab ? gab : gag;
    const float* bet  = isab ? bab : bag;
    const float* w    = isab ? fcw : agw;
    const float  bias = isab ? fcb[0] : agb[0];
    float acc = 0.f;
    for (int c = lane; c < 2 * DD; c += 32) {
        float v;
        if (c < DD) {
            v = x2[(size_t)row * DD + c];
        } else if (isab) {
            v = sab[(size_t)row * DD + c - DD];
        } else {
            v = xag[(size_t)(row - N_AB) * DD + c - DD];
        }
        float nv = (v - mean[c]) * rsqrtf(var[c] + 1e-5f) * gam[c] + bet[c];
        nv = fmaxf(nv, 0.f);
        acc += nv * w[c];
    }
#pragma unroll
    for (int off = 16; off; off >>= 1) acc += __shfl_xor(acc, off);
    if (lane == 0) out[row] = acc + bias;
}

// ---------------------------------------------------------------------------
extern "C" void kernel_launch(void* const* d_in, const int* in_sizes, int n_in,
                              void* d_out, int out_size, void* d_ws, size_t ws_size,
                              hipStream_t stream) {
    const float* sab  = (const float*)d_in[0];
    const float* xag  = (const float*)d_in[1];
    const float* W1   = (const float*)d_in[2];
    const float* as1  = (const float*)d_in[3];
    const float* ad1  = (const float*)d_in[4];
    const float* b1   = (const float*)d_in[5];
    const float* W2   = (const float*)d_in[6];
    const float* as2  = (const float*)d_in[7];
    const float* ad2  = (const float*)d_in[8];
    const float* b2   = (const float*)d_in[9];
    const float* bnabg = (const float*)d_in[10];
    const float* bnabb = (const float*)d_in[11];
    const float* bnagg = (const float*)d_in[12];
    const float* bnagb = (const float*)d_in[13];
    const float* fcw  = (const float*)d_in[14];
    const float* fcb  = (const float*)d_in[15];
    const float* agw  = (const float*)d_in[16];
    const float* agb  = (const float*)d_in[17];
    // d_in[18], d_in[19] = edge_src/edge_dst (block structure exploited analytically)

    float* ws   = (float*)d_ws;
    float* h    = ws;                       // NT*DD
    float* x1   = h   + (size_t)NT * DD;    // NT*DD
    float* x2   = x1  + (size_t)NT * DD;    // NT*DD
    float* als  = x2  + (size_t)NT * DD;    // NT
    float* ald  = als + NT;                 // NT
    float* mmax = ald + NT;                 // N_AB
    float* zs   = mmax + N_AB;              // N_AB
    float* mab  = zs + N_AB;                // 256
    float* vab  = mab + 256;                // 256
    float* mag  = vab + 256;                // 256
    float* vag  = mag + 256;                // 256

    const dim3 blk(256);

    // ---- layer 1 ----
    k_gemm_xw<<<NT / 16, blk, 0, stream>>>(sab, xag, W1, h);
    k_dots   <<<(NT + 7) / 8, blk, 0, stream>>>(h, as1, ad1, als, ald);
    k_ab_stats<<<N_AB / 8, blk, 0, stream>>>(als, ald, mmax, zs);
    k_ab_agg <<<BB * 2, blk, 0, stream>>>(h, als, ald, mmax, zs, b1, x1, 1);
    k_ag_agg <<<(BB * GG) / 16, blk, 0, stream>>>(h, als, ald, b1, x1, 1);

    // ---- layer 2 ----
    k_gemm_xw<<<NT / 16, blk, 0, stream>>>(x1, x1 + (size_t)N_AB * DD, W2, h);
    k_dots   <<<(NT + 7) / 8, blk, 0, stream>>>(h, as2, ad2, als, ald);
    k_ab_stats<<<N_AB / 8, blk, 0, stream>>>(als, ald, mmax, zs);
    k_ab_agg <<<BB * 2, blk, 0, stream>>>(h, als, ald, mmax, zs, b2, x2, 0);
    k_ag_agg <<<(BB * GG) / 16, blk, 0, stream>>>(h, als, ald, b2, x2, 0);

    // ---- heads ----
    k_bn_ab<<<1, blk, 0, stream>>>(x2, sab, mab, vab);
    k_bn_ag<<<256, blk, 0, stream>>>(x2, xag, mag, vag);
    k_head <<<(NT + 7) / 8, blk, 0, stream>>>(x2, sab, xag,
                                              mab, vab, bnabg, bnabb,
                                              mag, vag, bnagg, bnagb,
                                              fcw, fcb, agw, agb,
                                              (float*)d_out);
}